// MultiHeadAttentionLayer_26482768347806
// MI455X (gfx1250) — compile-verified
//
#include <hip/hip_runtime.h>

typedef __attribute__((ext_vector_type(2))) float v2f;
typedef __attribute__((ext_vector_type(8))) float v8f;

#define N_NODES 10000
#define N_EDGES 640000
#define D_IN    128
#define N_HEADS 8
#define D_HEAD  16
#define D_OUT   128   // N_HEADS * D_HEAD

__device__ __forceinline__ float clamp5(float x) {
    return fminf(fmaxf(x, -5.0f), 5.0f);
}

// ---------------- Kernel 0: zero the wV/z accumulators ----------------
__global__ void zero_ws(float* __restrict__ p, int n) {
    int i = blockIdx.x * blockDim.x + threadIdx.x;
    if (i < n) p[i] = 0.0f;
}

// ---------------- Kernel 1: node projections Qh/Kh/Vh -----------------
// grid (625, 24), block 32 (one wave). by>>3 selects matrix, (by&7)*16 = col.
// C = X(16x128) @ W(128x16) via 32 x V_WMMA_F32_16X16X4_F32.
__global__ void node_proj(const float* __restrict__ X,
                          const float* __restrict__ WQ,
                          const float* __restrict__ WK,
                          const float* __restrict__ WV,
                          float* __restrict__ Qh,
                          float* __restrict__ Kh,
                          float* __restrict__ Vh) {
    const int lane = threadIdx.x;
    const int half = lane >> 4;   // 0: K pair {0,1}, 1: K pair {2,3}
    const int l16  = lane & 15;
    const int rbase = blockIdx.x * 16;
    const int mat = blockIdx.y >> 3;
    const int col = (blockIdx.y & 7) * 16;
    const float* W  = (mat == 0) ? WQ : (mat == 1) ? WK : WV;
    float*      Out = (mat == 0) ? Qh : (mat == 1) ? Kh : Vh;

    v8f acc = {};
    const float* arow = X + (size_t)(rbase + l16) * D_IN + half * 2;
    const float* bcol = W + (size_t)(half * 2) * D_OUT + col + l16;
    for (int k = 0; k < D_IN; k += 4) {
        v2f a = { arow[0], arow[1] };          // contiguous b64 load
        v2f b = { bcol[0], bcol[D_OUT] };      // W[k][n], W[k+1][n]
        acc = __builtin_amdgcn_wmma_f32_16x16x4_f32(
            false, a, false, b, (short)0, acc, false, false);
        arow += 4;
        bcol += 4 * D_OUT;
    }
    // D layout: VGPR r -> row (r + half*8), lane l16 -> col n
    float* orow = Out + (size_t)(rbase + half * 8) * D_OUT + col + l16;
#pragma unroll
    for (int r = 0; r < 8; ++r)
        orow[(size_t)r * D_OUT] = acc[r];
}

// ---------------- Kernel 2: edge projection + score + scatter ---------
// grid (40000), block 256 = 8 wave32; wave h computes head h for 16 edges.
__global__ void edge_kernel(const float* __restrict__ EF,
                            const float* __restrict__ We,
                            const int*   __restrict__ src,
                            const int*   __restrict__ dst,
                            const float* __restrict__ Qh,
                            const float* __restrict__ Kh,
                            const float* __restrict__ Vh,
                            float* __restrict__ e_out,
                            float* __restrict__ wV,
                            float* __restrict__ z) {
    __shared__ float Atile[16 * D_IN];   // 8 KB edge-feature tile
    const int tid   = threadIdx.x;
    const int ebase = blockIdx.x * 16;

    // Cooperative 16x128 f32 tile load: 2048 floats = 512 float4, 256 threads x 2
    {
        const float4* g = (const float4*)(EF + (size_t)ebase * D_IN);
        float4* l = (float4*)Atile;
        l[tid]       = g[tid];
        l[tid + 256] = g[tid + 256];
    }
    __syncthreads();

    const int h    = tid >> 5;       // wave id == head
    const int lane = tid & 31;
    const int half = lane >> 4;
    const int l16  = lane & 15;
    const int col  = h * D_HEAD;

    // proj_e tile: (16 edges x 16 dims of head h)
    v8f acc = {};
    const float* a_lds = Atile + l16 * D_IN + half * 2;
    const float* bcol  = We + (size_t)(half * 2) * D_OUT + col + l16;
    for (int k = 0; k < D_IN; k += 4) {
        v2f a = { a_lds[k], a_lds[k + 1] };
        v2f b = { bcol[0], bcol[D_OUT] };
        acc = __builtin_amdgcn_wmma_f32_16x16x4_f32(
            false, a, false, b, (short)0, acc, false, false);
        bcol += 4 * D_OUT;
    }

    const float inv_sqrt_d = 0.25f;    // 1/sqrt(16)
    const int d = l16;                 // lane -> head-dim index
    // lane l16 caches indices for edge row l16; rows broadcast via shuffles
    const int mysrc = src[ebase + l16];
    const int mydst = dst[ebase + l16];

#pragma unroll
    for (int r = 0; r < 8; ++r) {
        const int rowlane = r + half * 8;        // C/D row held by this half
        const int s_n = __shfl(mysrc, rowlane, 32);
        const int d_n = __shfl(mydst, rowlane, 32);
        const int m   = ebase + rowlane;

        // L2-resident gathers of Kh[src], Qh[dst]
        float kq = Kh[(size_t)s_n * D_OUT + col + d] *
                   Qh[(size_t)d_n * D_OUT + col + d] * inv_sqrt_d;
        float sc = clamp5(kq) * acc[r];
        e_out[(size_t)m * D_OUT + col + d] = sc;

        // sum over d: butterfly across the 16-lane half (masks stay in-half)
        float t = sc;
        t += __shfl_xor(t, 1, 32);
        t += __shfl_xor(t, 2, 32);
        t += __shfl_xor(t, 4, 32);
        t += __shfl_xor(t, 8, 32);
        float s = expf(clamp5(t));

        // segment-sum scatter (L2 atomics; wV/z are L2-resident)
        atomicAdd(&wV[(size_t)d_n * D_OUT + col + d],
                  Vh[(size_t)s_n * D_OUT + col + d] * s);
        if (d == 0)
            atomicAdd(&z[(size_t)d_n * N_HEADS + h], s);
    }
}

// ---------------- Kernel 3: h_out = wV / (z + 1e-6) -------------------
__global__ void finalize(const float* __restrict__ wV,
                         const float* __restrict__ z,
                         float* __restrict__ h_out) {
    int i = blockIdx.x * blockDim.x + threadIdx.x;
    if (i < N_NODES * D_OUT)
        h_out[i] = wV[i] / (z[i >> 4] + 1e-6f);   // i>>4 == n*8 + h
}

extern "C" void kernel_launch(void* const* d_in, const int* in_sizes, int n_in,
                              void* d_out, int out_size, void* d_ws, size_t ws_size,
                              hipStream_t stream) {
    const float* node_feats = (const float*)d_in[0];
    const float* edge_feats = (const float*)d_in[1];
    const int*   src        = (const int*)d_in[2];
    const int*   dst        = (const int*)d_in[3];
    const float* WQ         = (const float*)d_in[4];
    const float* WK         = (const float*)d_in[5];
    const float* WV         = (const float*)d_in[6];
    const float* We         = (const float*)d_in[7];

    float* ws = (float*)d_ws;
    float* Qh = ws;
    float* Kh = Qh + (size_t)N_NODES * D_OUT;
    float* Vh = Kh + (size_t)N_NODES * D_OUT;
    float* wV = Vh + (size_t)N_NODES * D_OUT;
    float* z  = wV + (size_t)N_NODES * D_OUT;

    float* h_out = (float*)d_out;                          // [N, H, D]
    float* e_out = h_out + (size_t)N_NODES * D_OUT;        // [E, H, D]

    const int nacc = N_NODES * D_OUT + N_NODES * N_HEADS;  // wV + z contiguous
    zero_ws<<<(nacc + 255) / 256, 256, 0, stream>>>(wV, nacc);

    node_proj<<<dim3(N_NODES / 16, 24), 32, 0, stream>>>(
        node_feats, WQ, WK, WV, Qh, Kh, Vh);

    edge_kernel<<<dim3(N_EDGES / 16), 256, 0, stream>>>(
        edge_feats, We, src, dst, Qh, Kh, Vh, e_out, wV, z);

    finalize<<<(N_NODES * D_OUT + 255) / 256, 256, 0, stream>>>(wV, z, h_out);
}